// TinyMamba2Multi_44848048505424
// MI455X (gfx1250) — compile-verified
//
#include <hip/hip_runtime.h>
#include <math.h>

// ---- problem constants (from reference) ----
#define B_      64
#define L_      4096
#define DIN_    57
#define DM_     64
#define HD_     32
#define DS_     32
#define NH_     4
#define DINNER_ 128
#define CONV_DIM_ 192           // DINNER + 2*DS
#define DPROJ_  324             // 2*DINNER + 2*DS + NH
#define NPAD_   336             // DPROJ_ padded to 21*16
#define KCONV_  4
#define NOUT_   6
#define EPS_    1e-5f
#define M_      (B_ * L_)       // 262144 rows, multiple of 16

typedef float v2f __attribute__((ext_vector_type(2)));
typedef float v8f __attribute__((ext_vector_type(8)));

__device__ __forceinline__ float silu_f(float x) { return x / (1.f + __expf(-x)); }
__device__ __forceinline__ float softplus_f(float x) {
  return (x > 20.f) ? x : log1pf(__expf(x));
}

// V_WMMA_F32_16X16X4_F32 fragment mapping (CDNA5 ISA):
//   A 16x4 : lane<16 -> M=lane, VGPR0=K0,VGPR1=K1 ; lane>=16 -> same M, K2/K3
//   B  4x16: lane<16 -> N=lane, VGPR0=K0,VGPR1=K1 ; lane>=16 -> same N, K2/K3
//   C/D    : row m = i + 8*(lane>>4), col n = lane&15  (i = VGPR index 0..7)

// ---- GEMM 1: h0 = x @ W_in^T + b_in   (M x 57 -> M x 64) -------------------
// One wave owns one M-tile and ALL 4 N-tiles: A fragment loaded once per
// K-step and reused by 4 consecutive WMMAs.
__global__ void __launch_bounds__(256) k_gemm_in(const float* __restrict__ x,
                                                 const float* __restrict__ Win,
                                                 const float* __restrict__ bin,
                                                 float* __restrict__ h0)
{
  const int mt    = blockIdx.x * 8 + (threadIdx.x >> 5);   // M-tile per wave
  const int lane  = threadIdx.x & 31;
  const int mrow  = lane & 15;
  const int khalf = lane >> 4;
  const int nlo   = lane & 15;
  const float* arow = x + (size_t)(mt * 16 + mrow) * DIN_;

  const v8f vzero = {0.f, 0.f, 0.f, 0.f, 0.f, 0.f, 0.f, 0.f};
  v8f acc[4];
#pragma unroll
  for (int t = 0; t < 4; ++t) acc[t] = vzero;

  for (int k0 = 0; k0 < DIN_; k0 += 4) {
    const int k = k0 + 2 * khalf;
    v2f a;
    a.x = (k     < DIN_) ? arow[k]     : 0.f;
    a.y = (k + 1 < DIN_) ? arow[k + 1] : 0.f;
#pragma unroll
    for (int t = 0; t < 4; ++t) {
      const float* wrow = Win + (size_t)(t * 16 + nlo) * DIN_;
      v2f b;
      b.x = (k     < DIN_) ? wrow[k]     : 0.f;
      b.y = (k + 1 < DIN_) ? wrow[k + 1] : 0.f;
      acc[t] = __builtin_amdgcn_wmma_f32_16x16x4_f32(false, a, false, b,
                                                     (short)0, acc[t], false, false);
    }
  }

  const int mhalf = lane >> 4;
#pragma unroll
  for (int t = 0; t < 4; ++t) {
    const int n = t * 16 + nlo;
    const float bias = bin[n];
#pragma unroll
    for (int i = 0; i < 8; ++i)
      h0[(size_t)(mt * 16 + mhalf * 8 + i) * DM_ + n] = acc[t][i] + bias;
  }
}

// ---- GEMM 2: zxbcdt = h0 @ W_proj^T   (M x 64 -> M x 324) ------------------
// W_proj (324x64, 83KB) is staged once per block into LDS, transposed and
// packed as K-pairs so the per-K-step B fragment is one ds_load_b64.
// Each wave owns one M-tile and a strip of 7 N-tiles (21 total, 3 strips).
__global__ void __launch_bounds__(256) k_gemm_proj(const float* __restrict__ h0,
                                                   const float* __restrict__ Wp,
                                                   float* __restrict__ zx)
{
  __shared__ v2f Wt[(DM_ / 2) * NPAD_];                    // 32*336*8B = 84KB

  // cooperative staging: Wt[kpair][n] = (W[n][2k], W[n][2k+1]); pad n>=324 = 0
  for (int i = threadIdx.x; i < (DM_ / 2) * NPAD_; i += 256) {
    const int kh = i / NPAD_;
    const int n  = i % NPAD_;
    v2f w = {0.f, 0.f};
    if (n < DPROJ_) {
      w.x = Wp[(size_t)n * DM_ + 2 * kh];
      w.y = Wp[(size_t)n * DM_ + 2 * kh + 1];
    }
    Wt[kh * NPAD_ + n] = w;
  }
  __syncthreads();

  const int wid   = blockIdx.x * 8 + (threadIdx.x >> 5);
  const int mt    = wid / 3;
  const int g     = wid % 3;                               // N-strip (7 tiles)
  const int lane  = threadIdx.x & 31;
  const int mrow  = lane & 15;
  const int khalf = lane >> 4;
  const int nlo   = lane & 15;
  const float* arow = h0 + (size_t)(mt * 16 + mrow) * DM_;

  const v8f vzero = {0.f, 0.f, 0.f, 0.f, 0.f, 0.f, 0.f, 0.f};
  v8f acc[7];
#pragma unroll
  for (int t = 0; t < 7; ++t) acc[t] = vzero;

  for (int k0 = 0; k0 < DM_; k0 += 4) {
    const int k  = k0 + 2 * khalf;
    const int kh = k >> 1;
    v2f a;
    a.x = arow[k];
    a.y = arow[k + 1];
#pragma unroll
    for (int t = 0; t < 7; ++t) {
      const int ncol = (g * 7 + t) * 16 + nlo;             // < 336, zero-padded
      const v2f b = Wt[kh * NPAD_ + ncol];
      acc[t] = __builtin_amdgcn_wmma_f32_16x16x4_f32(false, a, false, b,
                                                     (short)0, acc[t], false, false);
    }
  }

  const int mhalf = lane >> 4;
#pragma unroll
  for (int t = 0; t < 7; ++t) {
    const int n = (g * 7 + t) * 16 + nlo;
    if (n < DPROJ_) {
#pragma unroll
      for (int i = 0; i < 8; ++i)
        zx[(size_t)(mt * 16 + mhalf * 8 + i) * DPROJ_ + n] = acc[t][i];
    }
  }
}

// ---- causal depthwise conv (K=4) + SiLU on xBC slice -----------------------
__global__ void k_conv_silu(const float* __restrict__ zx, const float* __restrict__ cw,
                            const float* __restrict__ cb, float* __restrict__ xc)
{
  const size_t idx = (size_t)blockIdx.x * blockDim.x + threadIdx.x;
  const int    c   = (int)(idx % CONV_DIM_);
  const size_t row = idx / CONV_DIM_;            // row = b*L + l
  const int    l   = (int)(row % L_);

  float acc = cb[c];
#pragma unroll
  for (int j = 0; j < KCONV_; ++j) {
    const int ll = l - (KCONV_ - 1) + j;
    if (ll >= 0)
      acc += cw[c * KCONV_ + j] * zx[(row + (size_t)(ll - l)) * DPROJ_ + DINNER_ + c];
  }
  xc[row * CONV_DIM_ + c] = silu_f(acc);
}

// ---- selective scan: one block per (b, head); 32 waves, wave = state row p,
//      lane = state col n; entire 32x32 state lives in one VGPR per lane -----
__global__ void __launch_bounds__(1024) k_scan(const float* __restrict__ zx,
                                               const float* __restrict__ xc,
                                               const float* __restrict__ dt_bias,
                                               const float* __restrict__ A_log,
                                               const float* __restrict__ Dp,
                                               float* __restrict__ y)
{
  const int head = blockIdx.x & (NH_ - 1);
  const int b    = blockIdx.x >> 2;
  const int p    = threadIdx.x >> 5;   // state row (HD), one wave per row
  const int n    = threadIdx.x & 31;   // state col (DS), one lane per col

  const float A_h = -__expf(A_log[head]);
  const float D_h = Dp[head];
  const float dtb = dt_bias[head];

  const float* xr = xc + (size_t)b * L_ * CONV_DIM_;
  const float* zr = zx + (size_t)b * L_ * DPROJ_;
  float*       yr = y  + (size_t)b * L_ * DINNER_;

  float hst = 0.f;
  for (int t = 0; t < L_; ++t) {
    const float* xrow = xr + (size_t)t * CONV_DIM_;
    if (p == 0 && t + 8 < L_)
      __builtin_prefetch(xrow + 8 * CONV_DIM_ + n * 6, 0, 1);  // global_prefetch_b8

    const float xh  = xrow[head * HD_ + p];
    const float Bn  = xrow[DINNER_ + n];
    const float Cn  = xrow[DINNER_ + DS_ + n];
    const float dtv = softplus_f(zr[(size_t)t * DPROJ_ + (DPROJ_ - NH_) + head] + dtb);
    const float dA  = __expf(dtv * A_h);

    hst = hst * dA + (dtv * xh) * Bn;

    float s = hst * Cn;
#pragma unroll
    for (int off = 16; off > 0; off >>= 1) s += __shfl_xor(s, off, 32);

    if (n == 0) yr[(size_t)t * DINNER_ + head * HD_ + p] = s + D_h * xh;
  }
}

// ---- gating + RMS norm, one wave per row (128 elems = 4/lane) --------------
__global__ void k_gate_rmsnorm(const float* __restrict__ zx, const float* __restrict__ nw,
                               float* __restrict__ y)
{
  const int row  = blockIdx.x * (blockDim.x >> 5) + (threadIdx.x >> 5);
  const int lane = threadIdx.x & 31;
  const float* zr = zx + (size_t)row * DPROJ_;
  float*       yr = y  + (size_t)row * DINNER_;

  float g[4];
  float ss = 0.f;
#pragma unroll
  for (int j = 0; j < 4; ++j) {
    const int d = lane + 32 * j;
    const float gv = yr[d] * silu_f(zr[d]);   // z is zxbcdt[..., :DINNER]
    g[j] = gv;
    ss += gv * gv;
  }
#pragma unroll
  for (int off = 16; off > 0; off >>= 1) ss += __shfl_xor(ss, off, 32);

  const float r = rsqrtf(ss * (1.f / DINNER_) + EPS_);
#pragma unroll
  for (int j = 0; j < 4; ++j) {
    const int d = lane + 32 * j;
    yr[d] = g[j] * r * nw[d];
  }
}

// ---- GEMM 3: h = h0 + gnorm @ W_out^T  (in place into h0) ------------------
// One wave owns one M-tile and all 4 N-tiles (K=128, no remainders).
__global__ void __launch_bounds__(256) k_gemm_out(const float* __restrict__ g,
                                                  const float* __restrict__ Wo,
                                                  float* __restrict__ h0)
{
  const int mt    = blockIdx.x * 8 + (threadIdx.x >> 5);
  const int lane  = threadIdx.x & 31;
  const int mrow  = lane & 15;
  const int khalf = lane >> 4;
  const int nlo   = lane & 15;
  const float* arow = g + (size_t)(mt * 16 + mrow) * DINNER_;

  const v8f vzero = {0.f, 0.f, 0.f, 0.f, 0.f, 0.f, 0.f, 0.f};
  v8f acc[4];
#pragma unroll
  for (int t = 0; t < 4; ++t) acc[t] = vzero;

  for (int k0 = 0; k0 < DINNER_; k0 += 4) {
    const int k = k0 + 2 * khalf;
    v2f a;
    a.x = arow[k];
    a.y = arow[k + 1];
#pragma unroll
    for (int t = 0; t < 4; ++t) {
      const float* wrow = Wo + (size_t)(t * 16 + nlo) * DINNER_;
      v2f b;
      b.x = wrow[k];
      b.y = wrow[k + 1];
      acc[t] = __builtin_amdgcn_wmma_f32_16x16x4_f32(false, a, false, b,
                                                     (short)0, acc[t], false, false);
    }
  }

  const int mhalf = lane >> 4;
#pragma unroll
  for (int t = 0; t < 4; ++t) {
    const int n = t * 16 + nlo;
#pragma unroll
    for (int i = 0; i < 8; ++i) {
      const size_t o = (size_t)(mt * 16 + mhalf * 8 + i) * DM_ + n;
      h0[o] += acc[t][i];      // residual add; each element owned by one lane
    }
  }
}

// ---- mean pool over L (deterministic LDS tree, no atomics) -----------------
__global__ void k_pool(const float* __restrict__ h, float* __restrict__ pooled)
{
  __shared__ float sm[512];
  const int b    = blockIdx.x;
  const int d    = threadIdx.x & 63;
  const int part = threadIdx.x >> 6;   // 8 partitions over L
  const float* hb = h + (size_t)b * L_ * DM_;

  float s = 0.f;
  for (int l = part; l < L_; l += 8) s += hb[(size_t)l * DM_ + d];
  sm[threadIdx.x] = s;
  __syncthreads();
  if (part == 0) {
    float tot = 0.f;
#pragma unroll
    for (int q = 0; q < 8; ++q) tot += sm[q * 64 + d];
    pooled[b * DM_ + d] = tot * (1.f / L_);
  }
}

// ---- classifier: out = pooled @ W_cls^T + b_cls  (64x6) --------------------
__global__ void k_cls(const float* __restrict__ pooled, const float* __restrict__ Wc,
                      const float* __restrict__ bc, float* __restrict__ out)
{
  const int t = threadIdx.x;
  if (t >= B_ * NOUT_) return;
  const int b = t / NOUT_, o = t % NOUT_;
  float s = bc[o];
  for (int d = 0; d < DM_; ++d) s += pooled[b * DM_ + d] * Wc[o * DM_ + d];
  out[t] = s;
}

// ---------------------------------------------------------------------------
extern "C" void kernel_launch(void* const* d_in, const int* in_sizes, int n_in,
                              void* d_out, int out_size, void* d_ws, size_t ws_size,
                              hipStream_t stream)
{
  (void)in_sizes; (void)n_in; (void)out_size; (void)ws_size;
  const float* x    = (const float*)d_in[0];
  const float* Win  = (const float*)d_in[1];
  const float* bin  = (const float*)d_in[2];
  const float* Wp   = (const float*)d_in[3];
  const float* cw   = (const float*)d_in[4];
  const float* cb   = (const float*)d_in[5];
  const float* dtb  = (const float*)d_in[6];
  const float* Alog = (const float*)d_in[7];
  const float* Dp   = (const float*)d_in[8];
  const float* nw   = (const float*)d_in[9];
  const float* Wo   = (const float*)d_in[10];
  const float* Wc   = (const float*)d_in[11];
  const float* bc   = (const float*)d_in[12];
  float* out = (float*)d_out;

  float* ws     = (float*)d_ws;
  float* h0     = ws;                                   // M * 64
  float* zx     = h0 + (size_t)M_ * DM_;                // M * 324
  float* xc     = zx + (size_t)M_ * DPROJ_;             // M * 192
  float* y      = xc + (size_t)M_ * CONV_DIM_;          // M * 128
  float* pooled = y  + (size_t)M_ * DINNER_;            // B * 64

  // GEMM1: 16384 M-tiles, 1 wave each (4 N-tiles per wave), 8 waves/block
  k_gemm_in<<<(M_ / 16) / 8, 256, 0, stream>>>(x, Win, bin, h0);
  // GEMM2: 16384 M-tiles x 3 N-strips = 49152 waves, 8 waves/block
  k_gemm_proj<<<(M_ / 16) * 3 / 8, 256, 0, stream>>>(h0, Wp, zx);
  // conv+silu over M*192 elements
  k_conv_silu<<<(unsigned)(((size_t)M_ * CONV_DIM_) / 256), 256, 0, stream>>>(zx, cw, cb, xc);
  // scan: 256 chains, 32 waves each
  k_scan<<<B_ * NH_, 1024, 0, stream>>>(zx, xc, dtb, Alog, Dp, y);
  // gate + rmsnorm: one wave per row
  k_gate_rmsnorm<<<M_ / 8, 256, 0, stream>>>(zx, nw, y);
  // GEMM3 + residual into h0
  k_gemm_out<<<(M_ / 16) / 8, 256, 0, stream>>>(y, Wo, h0);
  // pooling + classifier
  k_pool<<<B_, 512, 0, stream>>>(h0, pooled);
  k_cls<<<1, 384, 0, stream>>>(pooled, Wc, bc, out);
}